// DPS_41987600285727
// MI455X (gfx1250) — compile-verified
//
#include <hip/hip_runtime.h>

// Gumbel top-K straight-through estimator, algebraically reduced:
//   soft[n] = e[n] * P_{rank(n)},  P_r = sum_{k<=r} 1/D_k,  D_k = S - prefix(e_rank)
// One workgroup (1024 threads = 32 wave32) per batch row. Selection via
// 4096-bin radix histogram + exact rank counting among ~300 candidates.
// The two 256-length prefix sums run as f32 WMMA GEMMs with triangular
// ones-matrices (cumsum = X*U + Lstrict*(X*Ones)) on wave 0.
// Logits are staged into LDS via GLOBAL_LOAD_ASYNC_TO_LDS_B128 (ASYNCcnt path).

#define Bsz 32
#define Nn  8192
#define Kk  256
#define NBINS 4096
#define CAND_CAP 1024
#define TEMPER 1.0f
#define PEN  (-1000.0f)

#if defined(__has_builtin)
# if __has_builtin(__builtin_amdgcn_global_load_async_to_lds_b128) && \
     __has_builtin(__builtin_amdgcn_s_wait_asynccnt)
#  define HAVE_ASYNC_LDS 1
# endif
#endif
#ifndef HAVE_ASYNC_LDS
# define HAVE_ASYNC_LDS 0
#endif

typedef __attribute__((ext_vector_type(2))) float v2f;
typedef __attribute__((ext_vector_type(8))) float v8f;
#if HAVE_ASYNC_LDS
// Builtin signature (from hipcc diagnostic): params are pointers to
// __attribute__((vector_size(16))) int, in global (AS1) / LDS (AS3) spaces.
typedef int v4i_t __attribute__((vector_size(4 * sizeof(int))));
typedef __attribute__((address_space(1))) v4i_t GVec;
typedef __attribute__((address_space(3))) v4i_t LVec;
#endif

static __device__ inline v8f wmma4(v2f a, v2f b, v8f c) {
  // D(16x16) = A(16x4) * B(4x16) + C   (VOP3P V_WMMA_F32_16X16X4_F32)
  return __builtin_amdgcn_wmma_f32_16x16x4_f32(false, a, false, b, (short)0, c,
                                               false, false);
}

static __device__ inline unsigned flip_key(float f) {  // monotonic float->uint
  unsigned u = __float_as_uint(f);
  return (u & 0x80000000u) ? ~u : (u | 0x80000000u);
}
static __device__ inline float unflip_key(unsigned k) {
  unsigned u = (k & 0x80000000u) ? (k & 0x7FFFFFFFu) : ~k;
  return __uint_as_float(u);
}

// Inclusive cumsum of 256 f32 values (row-major 16x16 matrix X in LDS `x`)
// via 12 V_WMMA_F32_16X16X4_F32 ops.  `w` is a 256-f32 LDS scratch.
// Layout assumptions (wave32):
//   A 16x4 : lane&15 = M, VGPR{0,1} = K = 2*(lane>=16) + {0,1}
//   B 4x16 : lane&15 = N, VGPR{0,1} = K = 2*(lane>=16) + {0,1}
//   C/D    : lane&15 = N, VGPR r holds M = r + 8*(lane>=16)
static __device__ inline v8f cumsum256_wmma(const float* x, float* w, int lane) {
  const int half = lane >> 4;
  const int lj   = lane & 15;
  const int K0   = half * 2;
  v2f ones; ones.x = 1.0f; ones.y = 1.0f;

  // W = X * Ones  (W[i][j] = rowsum_i)
  v8f wacc = {0,0,0,0,0,0,0,0};
#pragma unroll
  for (int kb = 0; kb < 4; ++kb) {
    v2f a; a.x = x[16*lj + 4*kb + K0]; a.y = x[16*lj + 4*kb + K0 + 1];
    wacc = wmma4(a, ones, wacc);
  }
#pragma unroll
  for (int r = 0; r < 8; ++r) w[16*(r + 8*half) + lj] = wacc[r];  // D-layout spill
  __builtin_amdgcn_wave_barrier();  // wave-internal LDS ordering (DS in-order)

  // C2 = Lstrict * W  (carry of previous 16-chunks)
  v8f c2 = {0,0,0,0,0,0,0,0};
#pragma unroll
  for (int kb = 0; kb < 4; ++kb) {
    int k0 = 4*kb + K0;
    v2f a; a.x = (k0   < lj) ? 1.0f : 0.0f;   // Lstrict[M][k] = (k < M)
           a.y = (k0+1 < lj) ? 1.0f : 0.0f;
    v2f b; b.x = w[16*k0 + lj]; b.y = w[16*(k0+1) + lj];
    c2 = wmma4(a, b, c2);
  }

  // Y = X * U  (within-chunk inclusive prefix), U[m][j] = (m <= j)
  v8f y = {0,0,0,0,0,0,0,0};
#pragma unroll
  for (int kb = 0; kb < 4; ++kb) {
    int k0 = 4*kb + K0;
    v2f a; a.x = x[16*lj + k0]; a.y = x[16*lj + k0 + 1];
    v2f b; b.x = (k0   <= lj) ? 1.0f : 0.0f;
           b.y = (k0+1 <= lj) ? 1.0f : 0.0f;
    y = wmma4(a, b, y);
  }
#pragma unroll
  for (int r = 0; r < 8; ++r) y[r] += c2[r];
  return y;
}

__global__ __launch_bounds__(1024, 1) void DPS_gumbel_topk_st(
    const float* __restrict__ logits, const float* __restrict__ noise,
    const float* __restrict__ memin, float* __restrict__ out)
{
  __shared__ __align__(16) float sh_p[Nn];         // 32 KB perturbed row
  __shared__ unsigned sh_hist[NBINS];              // 16 KB, reused as u16 rank[Nn]
  __shared__ unsigned long long sh_cand[CAND_CAP]; // 8 KB candidate keys
  __shared__ float sh_selval[Kk];
  __shared__ float sh_x[Kk];
  __shared__ float sh_w[Kk];
  __shared__ float sh_P[Kk];
  __shared__ int sh_cnt[2];                        // [0]=threshold bucket, [1]=candCount
  __shared__ unsigned sh_keymax;
  __shared__ float sh_S;

  const int tid  = threadIdx.x;
  const int lane = tid & 31;
  const long rowOff = (long)blockIdx.x * Nn;
  const float invT = 1.0f / TEMPER;

  __builtin_prefetch(noise  + rowOff, 0, 1);   // global_prefetch_b8
  __builtin_prefetch(memin  + rowOff, 0, 1);

  // ---- phase 0: init ----
  for (int i = tid; i < NBINS; i += 1024) sh_hist[i] = 0u;
  if (tid == 0) { sh_cnt[0] = 0; sh_cnt[1] = 0; sh_keymax = 0u; sh_S = 0.0f; }

#if HAVE_ASYNC_LDS
  // Stage the logits row into LDS via the async DMA path (ASYNCcnt-tracked).
  // INST_OFFSET is added to BOTH global and LDS addresses (ISA 15.18.3), so
  // two calls with offset 0/16 move this thread's 32 contiguous bytes.
  {
    GVec* g = (GVec*)(logits + rowOff + tid * 8);
    LVec* l = (LVec*)&sh_p[tid * 8];
    __builtin_amdgcn_global_load_async_to_lds_b128(g, l, 0, 0);
    __builtin_amdgcn_global_load_async_to_lds_b128(g, l, 16, 0);
  }
#endif
  __syncthreads();

  // ---- phase 1: perturbed values, sortable keys, histogram, row max ----
  float p8[8]; unsigned key8[8]; float mm8[8];
  const float4* L4 = reinterpret_cast<const float4*>(logits + rowOff);
  const float4* N4 = reinterpret_cast<const float4*>(noise  + rowOff);
  const float4* M4 = reinterpret_cast<const float4*>(memin  + rowOff);
  (void)L4;
  float4 Lv[2], Nv[2], Mv[2];
  Nv[0] = N4[tid*2 + 0]; Nv[1] = N4[tid*2 + 1];   // overlap with async transfer
  Mv[0] = M4[tid*2 + 0]; Mv[1] = M4[tid*2 + 1];
#if HAVE_ASYNC_LDS
  __builtin_amdgcn_s_wait_asynccnt(0);
  __syncthreads();                                 // all waves' tiles visible
  Lv[0] = reinterpret_cast<const float4*>(sh_p)[tid*2 + 0];
  Lv[1] = reinterpret_cast<const float4*>(sh_p)[tid*2 + 1];
#else
  Lv[0] = L4[tid*2 + 0]; Lv[1] = L4[tid*2 + 1];
#endif

  unsigned kmax = 0u;
#pragma unroll
  for (int v = 0; v < 2; ++v) {
    float4 L = Lv[v], Nz = Nv[v], Mm = Mv[v];
    float pv[4] = { fmaf(PEN, Mm.x, L.x + Nz.x), fmaf(PEN, Mm.y, L.y + Nz.y),
                    fmaf(PEN, Mm.z, L.z + Nz.z), fmaf(PEN, Mm.w, L.w + Nz.w) };
    float mv[4] = { Mm.x, Mm.y, Mm.z, Mm.w };
#pragma unroll
    for (int q = 0; q < 4; ++q) {
      int e = v*4 + q;
      p8[e] = pv[q]; mm8[e] = mv[q];
      unsigned kk = flip_key(pv[q]);
      key8[e] = kk;
      kmax = kmax > kk ? kmax : kk;
      sh_p[tid*8 + e] = pv[q];
      atomicAdd(&sh_hist[kk >> 20], 1u);           // top 12 bits -> 4096 bins
    }
  }
#pragma unroll
  for (int d = 16; d > 0; d >>= 1) {
    unsigned o = __shfl_xor(kmax, d, 32);
    kmax = kmax > o ? kmax : o;
  }
  if (lane == 0) atomicMax(&sh_keymax, kmax);
  __syncthreads();

  // ---- phase 2: wave 0 finds the bucket containing the K-th largest key ----
  if (tid < 32) {
    unsigned s = 0;
    for (int b2 = 0; b2 < NBINS/32; ++b2) s += sh_hist[lane*(NBINS/32) + b2];
    unsigned t = s;                                // suffix sums over lanes
#pragma unroll
    for (int d = 1; d < 32; d <<= 1) {
      unsigned o = __shfl_down(t, d, 32);
      if (lane + d < 32) t += o;
    }
    unsigned long long ok = __ballot(t >= (unsigned)Kk);
    int lstar = 63 - __clzll(ok);                  // deepest region reaching >= K
    unsigned tnext = __shfl(t, (lstar + 1) & 31, 32);
    if (lane == 0) {
      unsigned cum = (lstar == 31) ? 0u : tnext;   // count strictly above region
      int Tb = lstar * (NBINS/32);
      for (int b2 = lstar*(NBINS/32) + (NBINS/32) - 1; b2 >= lstar*(NBINS/32); --b2) {
        unsigned h = sh_hist[b2];
        if (cum + h >= (unsigned)Kk) { Tb = b2; break; }
        cum += h;
      }
      sh_cnt[0] = Tb;
    }
  }
  __syncthreads();

  // ---- phase 3: candidate list, rank init, exp-sum S ----
  const float mrow = unflip_key(sh_keymax);
  const unsigned Tb = (unsigned)sh_cnt[0];
  unsigned short* rank16 = reinterpret_cast<unsigned short*>(sh_hist); // hist done
  float sexp = 0.0f;
#pragma unroll
  for (int e = 0; e < 8; ++e) {
    int n = tid*8 + e;
    rank16[n] = 0xFFFFu;
    sexp += __expf((p8[e] - mrow) * invT);
    if ((key8[e] >> 20) >= Tb) {
      int pos = atomicAdd(&sh_cnt[1], 1);
      if (pos < CAND_CAP)  // 45-bit key: value desc, then index asc (top_k order)
        sh_cand[pos] = ((unsigned long long)key8[e] << 32) | (unsigned)(Nn - 1 - n);
    }
  }
#pragma unroll
  for (int d = 16; d > 0; d >>= 1) sexp += __shfl_xor(sexp, d, 32);
  if (lane == 0) atomicAdd(&sh_S, sexp);
  __syncthreads();

  // ---- phase 4: exact ranks among candidates (LDS-broadcast compares) ----
  int M = sh_cnt[1]; if (M > CAND_CAP) M = CAND_CAP;
  for (int c = tid; c < M; c += 1024) {
    unsigned long long myk = sh_cand[c];
    int r = 0;
    for (int j = 0; j < M; ++j) r += (sh_cand[j] > myk) ? 1 : 0;
    if (r < Kk) {
      int n = Nn - 1 - (int)(unsigned)(myk & 0xFFFFFFFFull);
      sh_selval[r] = sh_p[n];
      rank16[n] = (unsigned short)r;
    }
  }
  __syncthreads();

  // ---- phase 5: WMMA prefix sums (wave 0, EXEC all ones) ----
  if (tid < 32) {
    for (int i = lane; i < Kk; i += 32)
      sh_x[i] = __expf((sh_selval[i] - mrow) * invT);   // e_rank
    __builtin_amdgcn_wave_barrier();
    v8f incE = cumsum256_wmma(sh_x, sh_w, lane);        // inclusive prefix of e_rank
    const float S = sh_S;
    const int half = lane >> 4, lj = lane & 15;
#pragma unroll
    for (int r = 0; r < 8; ++r) {
      int idx = 16*(r + 8*half) + lj;                   // D-layout element index
      float excl = incE[r] - sh_x[idx];                 // exclusive prefix
      float Dk = S - excl;                              // softmax denominator, D_0=S
      sh_x[idx] = 1.0f / Dk;
    }
    __builtin_amdgcn_wave_barrier();
    v8f P = cumsum256_wmma(sh_x, sh_w, lane);           // P_r = sum_{k<=r} 1/D_k
#pragma unroll
    for (int r = 0; r < 8; ++r) sh_P[16*(r + 8*half) + lj] = P[r];
  }
  __syncthreads();

  // ---- phase 6: output  out = (hard - soft) + soft ----
  float4* O4 = reinterpret_cast<float4*>(out + rowOff);
  const float Plast = sh_P[Kk - 1];
#pragma unroll
  for (int v = 0; v < 2; ++v) {
    float o[4];
#pragma unroll
    for (int q = 0; q < 4; ++q) {
      int e = v*4 + q;
      int n = tid*8 + e;
      float ev = __expf((p8[e] - mrow) * invT);
      unsigned r = rank16[n];
      float Pf = (r < (unsigned)Kk) ? sh_P[r] : Plast;
      float soft = ev * Pf;
      float hard = ((r < (unsigned)Kk) ? 1.0f : 0.0f) + mm8[e];
      o[q] = (hard - soft) + soft;
    }
    float4 ov; ov.x = o[0]; ov.y = o[1]; ov.z = o[2]; ov.w = o[3];
    O4[tid*2 + v] = ov;
  }
}

extern "C" void kernel_launch(void* const* d_in, const int* in_sizes, int n_in,
                              void* d_out, int out_size, void* d_ws, size_t ws_size,
                              hipStream_t stream) {
  (void)in_sizes; (void)n_in; (void)out_size; (void)d_ws; (void)ws_size;
  const float* logits = (const float*)d_in[0];
  const float* noise  = (const float*)d_in[1];
  const float* smem   = (const float*)d_in[2];
  float* outp = (float*)d_out;
  DPS_gumbel_topk_st<<<dim3(Bsz), dim3(1024), 0, stream>>>(logits, noise, smem, outp);
}